// GraphNeuralNetwork_25031069401543
// MI455X (gfx1250) — compile-verified
//
#include <hip/hip_runtime.h>
#include <hip/hip_bf16.h>

typedef float v2f __attribute__((ext_vector_type(2)));
typedef float v8f __attribute__((ext_vector_type(8)));

#define D 128
#define WPB 8          // waves per block (256 threads, wave32)
#define BLK_F 16384    // floats per 128x128 weight block

// ---------------------------------------------------------------------------
// helpers
// ---------------------------------------------------------------------------
__device__ __forceinline__ float half16_sum(float v) {
    v += __shfl_xor(v, 1);
    v += __shfl_xor(v, 2);
    v += __shfl_xor(v, 4);
    v += __shfl_xor(v, 8);
    return v;
}

// ---------------------------------------------------------------------------
// weight repack: permute a 128x128 row-major W into per-lane B-fragment order.
// For k-step s (4 k's), half h, lane-col m: the 16 floats lane (h,m) consumes
// (8 column tiles x 2 k's) are contiguous:
//   Wp[blk][ ((s*2+h)*16 + m)*16 + t*2 + j ] = W[blk][ (s*4+2h+j)*128 + t*16+m ]
// ---------------------------------------------------------------------------
__global__ void repack_k(const float* __restrict__ W, float* __restrict__ Wp, int nblk) {
    int i = blockIdx.x * blockDim.x + threadIdx.x;
    int total = nblk << 14;
    if (i >= total) return;
    int blk = i >> 14;
    int o = i & (BLK_F - 1);
    int j = o & 1;
    int t = (o >> 1) & 7;
    int m = (o >> 4) & 15;
    int h = (o >> 8) & 1;
    int s = o >> 9;
    int k = s * 4 + 2 * h + j;
    int col = t * 16 + m;
    Wp[i] = W[((size_t)blk << 14) + (size_t)k * D + col];
}

// ---------------------------------------------------------------------------
// degree / inverse-degree
// ---------------------------------------------------------------------------
__global__ void deg_count_k(const long long* __restrict__ ei, float* __restrict__ deg, int E) {
    int t = blockIdx.x * blockDim.x + threadIdx.x;
    if (t >= E) return;
    atomicAdd(&deg[ei[(size_t)E + t]], 1.0f);
}

__global__ void invdeg_k(float* __restrict__ deg, int N) {
    int t = blockIdx.x * blockDim.x + threadIdx.x;
    if (t >= N) return;
    deg[t] = 1.0f / fmaxf(deg[t], 1.0f);
}

// ---------------------------------------------------------------------------
// scatter: agg[dst] += node[src] * edge_attr   (32 threads / edge, float4 each)
// ---------------------------------------------------------------------------
__global__ __launch_bounds__(256)
void scatter_k(const float* __restrict__ node, const float* __restrict__ ea,
               const long long* __restrict__ ei, float* __restrict__ agg, int E) {
    long long t = (long long)blockIdx.x * blockDim.x + threadIdx.x;
    int e = (int)(t >> 5);
    if (e >= E) return;
    int f = ((int)t & 31) * 4;
    long long s = ei[e];
    long long d = ei[(size_t)E + e];
    float w = ea[e];
    const float4 v = *(const float4*)(node + (size_t)s * D + f);
    float* dst = agg + (size_t)d * D + f;
    atomicAdd(dst + 0, v.x * w);
    atomicAdd(dst + 1, v.y * w);
    atomicAdd(dst + 2, v.z * w);
    atomicAdd(dst + 3, v.w * w);
}

// ---------------------------------------------------------------------------
// WMMA helpers: 8 column tiles fed from 4 float4 B fragments
// ---------------------------------------------------------------------------
__device__ __forceinline__ void wmma8(v8f acc[8], const v2f& a, const float4 q[4]) {
#pragma unroll
    for (int t = 0; t < 8; ++t) {
        v2f b;
        if (t & 1) { b.x = q[t >> 1].z; b.y = q[t >> 1].w; }
        else       { b.x = q[t >> 1].x; b.y = q[t >> 1].y; }
        acc[t] = __builtin_amdgcn_wmma_f32_16x16x4_f32(
            false, a, false, b, (short)0, acc[t], false, false);
    }
}

// K=128 loop over two row tiles sharing B fragments
template <bool SCALE>
__device__ __forceinline__ void kloop_dual(const float* __restrict__ Xr0,
                                           const float* __restrict__ Xr1,
                                           float iv0, float iv1,
                                           const float* __restrict__ Wp,
                                           int half, int m,
                                           v8f acc0[8], v8f acc1[8]) {
    for (int s = 0; s < 32; ++s) {
        int k = s * 4 + 2 * half;
        v2f a0, a1;
        a0.x = Xr0[k]; a0.y = Xr0[k + 1];
        a1.x = Xr1[k]; a1.y = Xr1[k + 1];
        if (SCALE) { a0.x *= iv0; a0.y *= iv0; a1.x *= iv1; a1.y *= iv1; }
        const float4* bp = (const float4*)(Wp + (((s * 2 + half) * 16 + m) << 4));
        float4 q[4] = {bp[0], bp[1], bp[2], bp[3]};
        wmma8(acc0, a0, q);
        wmma8(acc1, a1, q);
    }
}

// ---------------------------------------------------------------------------
// epilogues (C layout: vgpr j holds row j + 8*half, col t*16 + m)
// ---------------------------------------------------------------------------
__device__ __forceinline__ void store_bias(v8f acc[8], const float* __restrict__ bias,
                                           float* __restrict__ out, int row0,
                                           int half, int m) {
#pragma unroll
    for (int t = 0; t < 8; ++t) {
        int col = t * 16 + m;
        float bv = bias[col];
#pragma unroll
        for (int j = 0; j < 8; ++j)
            out[(size_t)(row0 + j + 8 * half) * D + col] = acc[t][j] + bv;
    }
}

__device__ __forceinline__ void store_ln_relu(v8f acc[8], const float* __restrict__ bias,
                                              const float* __restrict__ g,
                                              const float* __restrict__ beta,
                                              float* __restrict__ out, int row0,
                                              int half, int m) {
#pragma unroll
    for (int t = 0; t < 8; ++t) {
        float bv = bias[t * 16 + m];
#pragma unroll
        for (int j = 0; j < 8; ++j) acc[t][j] += bv;
    }
    float mean[8], rstd[8];
#pragma unroll
    for (int j = 0; j < 8; ++j) {
        float s1 = 0.f, s2 = 0.f;
#pragma unroll
        for (int t = 0; t < 8; ++t) {
            float v = acc[t][j];
            s1 += v;
            s2 += v * v;
        }
        s1 = half16_sum(s1);
        s2 = half16_sum(s2);
        float mu = s1 * (1.0f / D);
        float var = s2 * (1.0f / D) - mu * mu;
        mean[j] = mu;
        rstd[j] = rsqrtf(var + 1e-5f);
    }
#pragma unroll
    for (int t = 0; t < 8; ++t) {
        int col = t * 16 + m;
        float gg = g[col], bb = beta[col];
#pragma unroll
        for (int j = 0; j < 8; ++j) {
            float v = (acc[t][j] - mean[j]) * rstd[j] * gg + bb;
            out[(size_t)(row0 + j + 8 * half) * D + col] = fmaxf(v, 0.0f);
        }
    }
}

// ---------------------------------------------------------------------------
// GraphConv: out = (agg*inv_deg) @ W_rel + b_rel + node @ W_root
// One wave -> 32 rows x 128 cols (two 16-row tiles), K = 256 in two phases.
// ---------------------------------------------------------------------------
__global__ __launch_bounds__(256)
void graphconv_k(const float* __restrict__ agg, const float* __restrict__ invd,
                 const float* __restrict__ node,
                 const float* __restrict__ WpRel, const float* __restrict__ brel,
                 const float* __restrict__ WpRoot,
                 float* __restrict__ out, int npairs, int N) {
    int p = blockIdx.x * WPB + (threadIdx.x >> 5);
    if (p >= npairs) return;                       // wave-uniform
    int lane = threadIdx.x & 31;
    int half = lane >> 4, m = lane & 15;
    int r0 = p * 32;
    bool has2 = (r0 + 16) < N;                     // wave-uniform
    int r1 = has2 ? r0 + 16 : r0;                  // clamp: loads stay in range

    float iv0 = invd[r0 + m];
    float iv1 = invd[r1 + m];
    const float* A0 = agg + (size_t)(r0 + m) * D;
    const float* A1 = agg + (size_t)(r1 + m) * D;
    const float* N0 = node + (size_t)(r0 + m) * D;
    const float* N1 = node + (size_t)(r1 + m) * D;

    v8f acc0[8] = {}, acc1[8] = {};
    kloop_dual<true>(A0, A1, iv0, iv1, WpRel, half, m, acc0, acc1);
    kloop_dual<false>(N0, N1, 0.f, 0.f, WpRoot, half, m, acc0, acc1);

    store_bias(acc0, brel, out, r0, half, m);
    if (has2) store_bias(acc1, brel, out, r1, half, m);
}

// ---------------------------------------------------------------------------
// MLP stage: out = relu(layer_norm(X @ W + bias) * g + beta), K = 128
// ---------------------------------------------------------------------------
__global__ __launch_bounds__(256)
void mlp_ln_relu_k(const float* __restrict__ X, const float* __restrict__ Wp,
                   const float* __restrict__ bias, const float* __restrict__ g,
                   const float* __restrict__ beta, float* __restrict__ out,
                   int npairs, int N) {
    int p = blockIdx.x * WPB + (threadIdx.x >> 5);
    if (p >= npairs) return;
    int lane = threadIdx.x & 31;
    int half = lane >> 4, m = lane & 15;
    int r0 = p * 32;
    bool has2 = (r0 + 16) < N;
    int r1 = has2 ? r0 + 16 : r0;

    const float* X0 = X + (size_t)(r0 + m) * D;
    const float* X1 = X + (size_t)(r1 + m) * D;

    v8f acc0[8] = {}, acc1[8] = {};
    kloop_dual<false>(X0, X1, 0.f, 0.f, Wp, half, m, acc0, acc1);

    store_ln_relu(acc0, bias, g, beta, out, r0, half, m);
    if (has2) store_ln_relu(acc1, bias, g, beta, out, r1, half, m);
}

// ---------------------------------------------------------------------------
// Jump-knowledge: out = [o0|o1|o2] @ jk_W + jk_b, K = 384 (three repacked blocks)
// ---------------------------------------------------------------------------
__global__ __launch_bounds__(256)
void jk_k(const float* __restrict__ o0, const float* __restrict__ o1,
          const float* __restrict__ o2, const float* __restrict__ WpJk,
          const float* __restrict__ jkb, float* __restrict__ out,
          int npairs, int N) {
    int p = blockIdx.x * WPB + (threadIdx.x >> 5);
    if (p >= npairs) return;
    int lane = threadIdx.x & 31;
    int half = lane >> 4, m = lane & 15;
    int r0 = p * 32;
    bool has2 = (r0 + 16) < N;
    int r1 = has2 ? r0 + 16 : r0;

    v8f acc0[8] = {}, acc1[8] = {};
    const float* srcs[3] = {o0, o1, o2};
#pragma unroll
    for (int l = 0; l < 3; ++l) {
        const float* X0 = srcs[l] + (size_t)(r0 + m) * D;
        const float* X1 = srcs[l] + (size_t)(r1 + m) * D;
        kloop_dual<false>(X0, X1, 0.f, 0.f, WpJk + (size_t)l * BLK_F, half, m, acc0, acc1);
    }

    store_bias(acc0, jkb, out, r0, half, m);
    if (has2) store_bias(acc1, jkb, out, r1, half, m);
}

// ---------------------------------------------------------------------------
// host orchestration
// ---------------------------------------------------------------------------
extern "C" void kernel_launch(void* const* d_in, const int* in_sizes, int n_in,
                              void* d_out, int out_size, void* d_ws, size_t ws_size,
                              hipStream_t stream) {
    const float* node      = (const float*)d_in[0];
    const float* edge_attr = (const float*)d_in[1];
    const float* W_rel     = (const float*)d_in[2];
    const float* b_rel     = (const float*)d_in[3];
    const float* W_root    = (const float*)d_in[4];
    const float* ln1_g     = (const float*)d_in[5];
    const float* ln1_b     = (const float*)d_in[6];
    const float* ln2_g     = (const float*)d_in[7];
    const float* ln2_b     = (const float*)d_in[8];
    const float* W1        = (const float*)d_in[9];
    const float* b1        = (const float*)d_in[10];
    const float* W2        = (const float*)d_in[11];
    const float* b2        = (const float*)d_in[12];
    const float* jkW       = (const float*)d_in[13];
    const float* jkb       = (const float*)d_in[14];
    const long long* ei    = (const long long*)d_in[15];

    const int N = in_sizes[0] / D;
    const int E = in_sizes[1];
    const size_t nd = (size_t)N * D;

    float* ws  = (float*)d_ws;
    float* agg = ws;            // N*D   (also reused as MLP1 output buffer)
    float* tmp = agg + nd;      // N*D
    float* o0  = tmp + nd;      // N*D
    float* o1  = o0 + nd;       // N*D
    float* o2  = o1 + nd;       // N*D
    float* deg = o2 + nd;       // N     (becomes inv_deg)
    float* wpRel  = deg + N;            // 3 * 16K
    float* wpRoot = wpRel  + 3 * BLK_F; // 3 * 16K
    float* wpW1   = wpRoot + 3 * BLK_F; // 3 * 16K
    float* wpW2   = wpW1   + 3 * BLK_F; // 3 * 16K
    float* wpJk   = wpW2   + 3 * BLK_F; // 3 * 16K
    float* outs[3] = {o0, o1, o2};

    // swizzle-repack all weights into WMMA B-fragment order (~1 MB total)
    const int rblocks = (3 * BLK_F + 255) / 256;
    repack_k<<<rblocks, 256, 0, stream>>>(W_rel,  wpRel,  3);
    repack_k<<<rblocks, 256, 0, stream>>>(W_root, wpRoot, 3);
    repack_k<<<rblocks, 256, 0, stream>>>(W1,     wpW1,   3);
    repack_k<<<rblocks, 256, 0, stream>>>(W2,     wpW2,   3);
    repack_k<<<rblocks, 256, 0, stream>>>(jkW,    wpJk,   3);

    // degree -> inv_deg
    hipMemsetAsync(deg, 0, (size_t)N * sizeof(float), stream);
    deg_count_k<<<(E + 255) / 256, 256, 0, stream>>>(ei, deg, E);
    invdeg_k<<<(N + 255) / 256, 256, 0, stream>>>(deg, N);

    const int ntiles16 = (N + 15) / 16;
    const int npairs   = (ntiles16 + 1) / 2;
    const int gblocks  = (npairs + WPB - 1) / WPB;
    const long long sthreads = (long long)E * 32;
    const int sblocks = (int)((sthreads + 255) / 256);

    const float* cur = node;
    for (int l = 0; l < 3; ++l) {
        hipMemsetAsync(agg, 0, nd * sizeof(float), stream);
        scatter_k<<<sblocks, 256, 0, stream>>>(cur, edge_attr, ei, agg, E);
        graphconv_k<<<gblocks, 256, 0, stream>>>(
            agg, deg, cur,
            wpRel + (size_t)l * BLK_F, b_rel + (size_t)l * D,
            wpRoot + (size_t)l * BLK_F, tmp, npairs, N);
        mlp_ln_relu_k<<<gblocks, 256, 0, stream>>>(
            tmp, wpW1 + (size_t)l * BLK_F, b1 + (size_t)l * D,
            ln1_g + (size_t)l * D, ln1_b + (size_t)l * D, agg, npairs, N);
        mlp_ln_relu_k<<<gblocks, 256, 0, stream>>>(
            agg, wpW2 + (size_t)l * BLK_F, b2 + (size_t)l * D,
            ln2_g + (size_t)l * D, ln2_b + (size_t)l * D, outs[l], npairs, N);
        cur = outs[l];
    }

    jk_k<<<gblocks, 256, 0, stream>>>(o0, o1, o2, wpJk, jkb, (float*)d_out, npairs, N);
}